// GNNModel_88278757802580
// MI455X (gfx1250) — compile-verified
//
#include <hip/hip_runtime.h>
#include <hip/hip_bf16.h>
#include <math.h>

typedef float v2f __attribute__((ext_vector_type(2)));
typedef float v8f __attribute__((ext_vector_type(8)));

#define DD 64
#define GG 64
#define NCLS 2
#define LDST 66   // padded LDS row stride (even -> 8B-aligned b64 ds loads, no 16-way bank conflict)

// ---------------- degree / normalization ----------------
__global__ __launch_bounds__(256) void k_deg_init(float* deg, int n) {
    int i = blockIdx.x * 256 + threadIdx.x;
    if (i < n) deg[i] = 1.0f;                     // self loop
}

__global__ __launch_bounds__(256) void k_deg_count(const int* __restrict__ ei, float* deg, int E) {
    int e = blockIdx.x * 256 + threadIdx.x;
    if (e < E) unsafeAtomicAdd(&deg[ei[E + e]], 1.0f);   // dst row of edge_index
}

__global__ __launch_bounds__(256) void k_rsqrt(float* deg, int n) {
    int i = blockIdx.x * 256 + threadIdx.x;
    if (i < n) deg[i] = rsqrtf(deg[i]);           // deg >= 1 always
}

// per-edge symmetric norm, computed once, reused by both layers' scatters
__global__ __launch_bounds__(256) void k_norm(const int* __restrict__ ei,
                                              const float* __restrict__ dinv,
                                              float* __restrict__ norm, int E) {
    int e = blockIdx.x * 256 + threadIdx.x;
    if (e < E) norm[e] = dinv[ei[e]] * dinv[ei[E + e]];
}

// ---------------- GEMM: C[n,64] = A[n,64] @ W[64,64] via V_WMMA_F32_16X16X4_F32 ----------------
__global__ __launch_bounds__(256) void k_gemm64(const float* __restrict__ A,
                                                const float* __restrict__ W,
                                                float* __restrict__ C, int nrows) {
    __shared__ float WT[DD * LDST];               // W transposed: WT[n*LDST + k] = W[k*64 + n]
    int tid = threadIdx.x;
    for (int i = tid; i < DD * DD; i += 256) {
        int k = i >> 6, n = i & 63;
        WT[n * LDST + k] = W[i];
    }
    __syncthreads();

    int wave = tid >> 5;                          // 8 waves / block
    int lane = tid & 31;
    int r0 = blockIdx.x * 128 + wave * 16;        // 16 rows per wave
    if (r0 >= nrows) return;                      // wave-uniform: EXEC stays all-ones

    int m  = lane & 15;
    int hi = lane >> 4;                           // half-wave selector
    int ar = r0 + m; if (ar > nrows - 1) ar = nrows - 1;   // clamp, keep EXEC full
    const float* ap = A + (size_t)ar * DD;

    v8f acc0 = {}, acc1 = {}, acc2 = {}, acc3 = {};
#pragma unroll
    for (int k = 0; k < DD; k += 4) {
        int kk = k + 2 * hi;
        v2f a  = *(const v2f*)(ap + kk);                              // A: M=lane%16, K=kk,kk+1
        v2f b0 = *(const v2f*)(&WT[(0 * 16 + m) * LDST + kk]);        // B: N=lane%16, K=kk,kk+1
        v2f b1 = *(const v2f*)(&WT[(1 * 16 + m) * LDST + kk]);
        v2f b2 = *(const v2f*)(&WT[(2 * 16 + m) * LDST + kk]);
        v2f b3 = *(const v2f*)(&WT[(3 * 16 + m) * LDST + kk]);
        acc0 = __builtin_amdgcn_wmma_f32_16x16x4_f32(false, a, false, b0, (short)0, acc0, false, false);
        acc1 = __builtin_amdgcn_wmma_f32_16x16x4_f32(false, a, false, b1, (short)0, acc1, false, false);
        acc2 = __builtin_amdgcn_wmma_f32_16x16x4_f32(false, a, false, b2, (short)0, acc2, false, false);
        acc3 = __builtin_amdgcn_wmma_f32_16x16x4_f32(false, a, false, b3, (short)0, acc3, false, false);
    }

#pragma unroll
    for (int j = 0; j < 8; ++j) {                 // C: VGPR j -> row j + 8*hi, col = tile*16 + m
        int row = r0 + j + 8 * hi;
        if (row < nrows) {
            float* cp = C + (size_t)row * DD + m;
            cp[0]  = acc0[j];
            cp[16] = acc1[j];
            cp[32] = acc2[j];
            cp[48] = acc3[j];
        }
    }
}

// ---------------- edge scatter: agg[dst] += xw[src] * norm[e] ----------------
__global__ __launch_bounds__(256) void k_scatter(const int* __restrict__ ei,
                                                 const float* __restrict__ xw,
                                                 const float* __restrict__ norm,
                                                 float* __restrict__ agg, int E) {
    int gid = blockIdx.x * 256 + threadIdx.x;
    int e = gid >> 4;                              // 16 lanes per edge
    if (e >= E) return;
    int c = (gid & 15) << 2;                       // float4 chunk
    int s = ei[e];
    int d = ei[E + e];
    float w = norm[e];
    float4 v = *(const float4*)(xw + (size_t)s * DD + c);
    float* p = agg + (size_t)d * DD + c;
    unsafeAtomicAdd(p + 0, v.x * w);
    unsafeAtomicAdd(p + 1, v.y * w);
    unsafeAtomicAdd(p + 2, v.z * w);
    unsafeAtomicAdd(p + 3, v.w * w);
}

// ---------------- layer-1 epilogue: self-loop + bias + ReLU (in place on agg) ----------------
__global__ __launch_bounds__(256) void k_finish(float* __restrict__ agg,
                                                const float* __restrict__ xw,
                                                const float* __restrict__ dinv,
                                                const float* __restrict__ bias, int n) {
    int gid = blockIdx.x * 256 + threadIdx.x;
    int i = gid >> 4;
    if (i >= n) return;
    int c = (gid & 15) << 2;
    float w = dinv[i]; w *= w;                     // self-loop norm = dinv^2
    float4 av = *(const float4*)(agg + (size_t)i * DD + c);
    float4 xv = *(const float4*)(xw + (size_t)i * DD + c);
    float4 bv = *(const float4*)(bias + c);
    float4 r;
    r.x = fmaxf(av.x + xv.x * w + bv.x, 0.0f);
    r.y = fmaxf(av.y + xv.y * w + bv.y, 0.0f);
    r.z = fmaxf(av.z + xv.z * w + bv.z, 0.0f);
    r.w = fmaxf(av.w + xv.w * w + bv.w, 0.0f);
    *(float4*)(agg + (size_t)i * DD + c) = r;
}

// ---------------- layer-2 epilogue fused with mean-pool scatter ----------------
// h = relu(agg + xw*dinv^2 + b2) is consumed only by pooling -> never materialize it.
__global__ __launch_bounds__(256) void k_finish_pool(const float* __restrict__ agg,
                                                     const float* __restrict__ xw,
                                                     const float* __restrict__ dinv,
                                                     const float* __restrict__ bias,
                                                     const int* __restrict__ batch,
                                                     float* __restrict__ pool,
                                                     float* __restrict__ cnt, int n) {
    int gid = blockIdx.x * 256 + threadIdx.x;
    int i = gid >> 4;
    if (i >= n) return;
    int c = (gid & 15) << 2;
    float w = dinv[i]; w *= w;
    float4 av = *(const float4*)(agg + (size_t)i * DD + c);
    float4 xv = *(const float4*)(xw + (size_t)i * DD + c);
    float4 bv = *(const float4*)(bias + c);
    int g = batch[i];
    float* p = pool + g * DD + c;
    unsafeAtomicAdd(p + 0, fmaxf(av.x + xv.x * w + bv.x, 0.0f));
    unsafeAtomicAdd(p + 1, fmaxf(av.y + xv.y * w + bv.y, 0.0f));
    unsafeAtomicAdd(p + 2, fmaxf(av.z + xv.z * w + bv.z, 0.0f));
    unsafeAtomicAdd(p + 3, fmaxf(av.w + xv.w * w + bv.w, 0.0f));
    if (c == 0) unsafeAtomicAdd(&cnt[g], 1.0f);
}

// ---------------- FC head + log_softmax ----------------
__global__ __launch_bounds__(64) void k_head(const float* __restrict__ pool,
                                             const float* __restrict__ cnt,
                                             const float* __restrict__ Wfc,
                                             const float* __restrict__ bfc,
                                             float* __restrict__ out) {
    int g = threadIdx.x;
    if (g >= GG) return;
    float inv = 1.0f / fmaxf(cnt[g], 1.0f);
    float l0 = bfc[0], l1 = bfc[1];
#pragma unroll 8
    for (int j = 0; j < DD; ++j) {
        float p = pool[g * DD + j] * inv;
        l0 += p * Wfc[j * NCLS + 0];
        l1 += p * Wfc[j * NCLS + 1];
    }
    float mx = fmaxf(l0, l1);
    float lse = mx + logf(expf(l0 - mx) + expf(l1 - mx));
    out[g * NCLS + 0] = l0 - lse;
    out[g * NCLS + 1] = l1 - lse;
}

extern "C" void kernel_launch(void* const* d_in, const int* in_sizes, int n_in,
                              void* d_out, int out_size, void* d_ws, size_t ws_size,
                              hipStream_t stream) {
    const float* x    = (const float*)d_in[0];
    const int*   ei   = (const int*)d_in[1];   // [2,E] flat: row0 = src, row1 = dst
    const int*   bat  = (const int*)d_in[2];
    const float* W1   = (const float*)d_in[3];
    const float* b1   = (const float*)d_in[4];
    const float* W2   = (const float*)d_in[5];
    const float* b2   = (const float*)d_in[6];
    const float* Wfc  = (const float*)d_in[7];
    const float* bfc  = (const float*)d_in[8];
    float* out = (float*)d_out;

    int N_ = in_sizes[0] / DD;     // 100000
    int E_ = in_sizes[1] / 2;      // 1600000

    // workspace: buf0 (xw), buf1 (agg/h), dinv, norm, pool, cnt  (~58 MB)
    float* buf0 = (float*)d_ws;
    float* buf1 = buf0 + (size_t)N_ * DD;
    float* dinv = buf1 + (size_t)N_ * DD;
    float* norm = dinv + N_;
    float* pool = norm + E_;
    float* cnt  = pool + GG * DD;

    int nb = (N_ + 255) / 256;
    int eb = (E_ + 255) / 256;
    int sb = (int)(((size_t)E_ * 16 + 255) / 256);
    int fb = (int)(((size_t)N_ * 16 + 255) / 256);
    int gb = (N_ + 127) / 128;

    // symmetric normalization: dinv[i], then per-edge norm
    k_deg_init<<<nb, 256, 0, stream>>>(dinv, N_);
    k_deg_count<<<eb, 256, 0, stream>>>(ei, dinv, E_);
    k_rsqrt<<<nb, 256, 0, stream>>>(dinv, N_);
    k_norm<<<eb, 256, 0, stream>>>(ei, dinv, norm, E_);

    // layer 1: xw = x@W1 ; agg = scatter ; h = relu(agg + xw*dinv^2 + b1)
    k_gemm64<<<gb, 256, 0, stream>>>(x, W1, buf0, N_);
    hipMemsetAsync(buf1, 0, (size_t)N_ * DD * sizeof(float), stream);
    k_scatter<<<sb, 256, 0, stream>>>(ei, buf0, norm, buf1, E_);
    k_finish<<<fb, 256, 0, stream>>>(buf1, buf0, dinv, b1, N_);

    // layer 2: xw2 = h@W2 ; agg2 = scatter ; epilogue fused with mean-pool
    k_gemm64<<<gb, 256, 0, stream>>>(buf1, W2, buf0, N_);
    hipMemsetAsync(buf1, 0, (size_t)N_ * DD * sizeof(float), stream);
    hipMemsetAsync(pool, 0, (size_t)(GG * DD + GG) * sizeof(float), stream);
    k_scatter<<<sb, 256, 0, stream>>>(ei, buf0, norm, buf1, E_);
    k_finish_pool<<<fb, 256, 0, stream>>>(buf1, buf0, dinv, b2, bat, pool, cnt, N_);

    // head
    k_head<<<1, 64, 0, stream>>>(pool, cnt, Wfc, bfc, out);
}